// MambaUNet_79336635892032
// MI455X (gfx1250) — compile-verified
//
#include <hip/hip_runtime.h>
#include <math.h>

typedef float v2f __attribute__((ext_vector_type(2)));
typedef float v8f __attribute__((ext_vector_type(8)));

#define D_MODEL 96
#define D_STATE 16
#define D_INNER 192
#define DT_RANK 6
#define KDIRS   4
#define BATCH   4
#define HH      64
#define WWID    64
#define LSEQ    4096                 // 64*64
#define NPIX    (BATCH*LSEQ)         // 16384
#define NDBL    (DT_RANK + 2*D_STATE) // 38
#define NSEQ    (BATCH*KDIRS*D_INNER) // 3072 independent scan sequences
#define NCH     16                    // scan chunks per sequence
#define CHL     (LSEQ/NCH)            // 256 steps per chunk

// ---------------------------------------------------------------------------
// fp32 WMMA 16x16x4 helper.
// A fragment (16x4, 32-bit): lanes 0-15 hold M=lane, K={k,k+1}; lanes 16-31
// hold M=lane-16, K={k+2,k+3}.  B fragment (4x16): vgpr0 = row K (k or k+2
// by lane half), vgpr1 = row K+1, N = lane&15.  C/D: vgpr i -> M=i (+8 for
// upper half), N = lane&15.
// ---------------------------------------------------------------------------
__device__ __forceinline__ v8f wmma4(v8f acc, v2f a, v2f b) {
  return __builtin_amdgcn_wmma_f32_16x16x4_f32(
      /*neg_a=*/false, a, /*neg_b=*/false, b,
      /*c_mod=*/(short)0, acc, /*reuse_a=*/false, /*reuse_b=*/false);
}

__device__ __forceinline__ float sigmoid_f(float x) {
  return 1.0f / (1.0f + __expf(-x));
}

// ---------------------------------------------------------------------------
// Kernel 1: in_proj GEMM.  xz[m][e] = sum_c x[m][c] * W[e][c]
// M = 16384 pixels, N = 384 (-> xp / z split), K = 96.  One wave per tile.
// ---------------------------------------------------------------------------
__global__ void k_in_proj(const float* __restrict__ x, const float* __restrict__ w,
                          float* __restrict__ xp, float* __restrict__ z) {
  const int lane = threadIdx.x & 31;
  const int half = lane >> 4;
  const int id   = lane & 15;
  int wave = (blockIdx.x * blockDim.x + threadIdx.x) >> 5;
  const int ntiles = (2 * D_INNER) / 16;          // 24
  const int mtile = wave / ntiles;
  const int ntile = wave % ntiles;
  const int mbase = mtile * 16, nbase = ntile * 16;

  const float* Arow = x + (size_t)(mbase + id) * D_MODEL;
  const float* Brow = w + (size_t)(nbase + id) * D_MODEL;   // W is (384,96) row-major
  v8f acc = {};
  for (int k = 0; k < D_MODEL; k += 4) {
    const int ka = k + 2 * half;
    v2f a, b;
    a.x = Arow[ka];     a.y = Arow[ka + 1];
    b.x = Brow[ka];     b.y = Brow[ka + 1];
    acc = wmma4(acc, a, b);
  }
  const int e = nbase + id;
  #pragma unroll
  for (int i = 0; i < 8; ++i) {
    const int m = mbase + i + 8 * half;
    const float v = acc[i];
    if (e < D_INNER) xp[(size_t)m * D_INNER + e] = v;
    else             z [(size_t)m * D_INNER + (e - D_INNER)] = v;
  }
}

// ---------------------------------------------------------------------------
// Kernel 2: depthwise 3x3 conv + bias + SiLU, write both directional copies.
// xs2 layout: (B, dir{HW,WH}, D_INNER, L) with l contiguous.
// ---------------------------------------------------------------------------
__global__ void k_conv(const float* __restrict__ xp, const float* __restrict__ cw,
                       const float* __restrict__ cb, float* __restrict__ xs2) {
  const int tid = blockIdx.x * blockDim.x + threadIdx.x;
  const int d   = tid % D_INNER;
  const int pix = tid / D_INNER;
  const int b   = pix / LSEQ;
  const int l0  = pix % LSEQ;
  const int h = l0 / WWID, w = l0 % WWID;

  float s = cb[d];
  #pragma unroll
  for (int i = 0; i < 3; ++i) {
    const int hh = h + i - 1;
    if (hh < 0 || hh >= HH) continue;
    #pragma unroll
    for (int j = 0; j < 3; ++j) {
      const int ww = w + j - 1;
      if (ww < 0 || ww >= WWID) continue;
      s += xp[((size_t)b * LSEQ + hh * WWID + ww) * D_INNER + d] * cw[d * 9 + i * 3 + j];
    }
  }
  const float v = s * sigmoid_f(s);            // SiLU
  const int l1 = w * HH + h;                   // WH (transposed) order
  xs2[(((size_t)b * 2 + 0) * D_INNER + d) * LSEQ + l0] = v;
  xs2[(((size_t)b * 2 + 1) * D_INNER + d) * LSEQ + l1] = v;
}

// ---------------------------------------------------------------------------
// Kernel 3: x_proj per direction.  x_dbl[b,k,c,l] = sum_d W_k[c][d] * xs[b,k,d,l]
// WMMA: M = l (4096), N = c (38 -> 3 tiles, guarded), K = 192.
// Flipped directions (k>=2) read A at position L-1-l.
// ---------------------------------------------------------------------------
__global__ void k_xproj(const float* __restrict__ xs2, const float* __restrict__ xpw,
                        float* __restrict__ xdbl) {
  const int lane = threadIdx.x & 31;
  const int half = lane >> 4, id = lane & 15;
  int wave = (blockIdx.x * blockDim.x + threadIdx.x) >> 5;
  const int ntile = wave % 3;  wave /= 3;
  const int mtile = wave % 256; wave /= 256;
  const int bk = wave;                 // 0..15
  const int k = bk % 4, b = bk / 4;
  const int dir = k & 1, flip = k >> 1;
  const int mbase = mtile * 16, nbase = ntile * 16;

  const float* xsb = xs2 + ((size_t)(b * 2 + dir)) * D_INNER * LSEQ;
  const int m   = mbase + id;
  const int pos = flip ? (LSEQ - 1 - m) : m;
  const int n   = nbase + id;
  const bool nv = (n < NDBL);
  const float* Wrow = xpw + (size_t)(k * NDBL + (nv ? n : 0)) * D_INNER;

  v8f acc = {};
  for (int kk = 0; kk < D_INNER; kk += 4) {
    const int ka = kk + 2 * half;
    v2f a, bb;
    a.x  = xsb[(size_t)ka * LSEQ + pos];
    a.y  = xsb[(size_t)(ka + 1) * LSEQ + pos];
    bb.x = nv ? Wrow[ka]     : 0.0f;
    bb.y = nv ? Wrow[ka + 1] : 0.0f;
    acc = wmma4(acc, a, bb);
  }
  if (nv) {
    float* out = xdbl + ((size_t)(bk * NDBL + n)) * LSEQ + mbase;
    #pragma unroll
    for (int i = 0; i < 8; ++i) out[i + 8 * half] = acc[i];
  }
}

// ---------------------------------------------------------------------------
// Kernel 4: dt_proj + softplus.
// delta[b,k,d,l] = softplus( sum_r dtw[k,d,r] * dts[b,k,r,l] + bias[k,d] )
// WMMA: M = l, N = d (192 -> 12 tiles), K = 6 padded to 8 (guarded).
// ---------------------------------------------------------------------------
__global__ void k_dtproj(const float* __restrict__ xdbl, const float* __restrict__ dtw,
                         const float* __restrict__ dtb, float* __restrict__ delta) {
  const int lane = threadIdx.x & 31;
  const int half = lane >> 4, id = lane & 15;
  int wave = (blockIdx.x * blockDim.x + threadIdx.x) >> 5;
  const int ntile = wave % 12;  wave /= 12;
  const int mtile = wave % 256; wave /= 256;
  const int bk = wave;
  const int k  = bk % 4;
  const int mbase = mtile * 16, nbase = ntile * 16;
  const int n = nbase + id;                       // d index

  const float* Abase = xdbl + (size_t)bk * NDBL * LSEQ;   // rows 0..5 are dts
  const float* Wrow  = dtw + (size_t)(k * D_INNER + n) * DT_RANK;
  v8f acc = {};
  #pragma unroll
  for (int kk = 0; kk < 8; kk += 4) {
    const int ka = kk + 2 * half;
    v2f a, bb;
    a.x  = (ka     < DT_RANK) ? Abase[(size_t)ka * LSEQ + mbase + id]       : 0.0f;
    a.y  = (ka + 1 < DT_RANK) ? Abase[(size_t)(ka + 1) * LSEQ + mbase + id] : 0.0f;
    bb.x = (ka     < DT_RANK) ? Wrow[ka]     : 0.0f;
    bb.y = (ka + 1 < DT_RANK) ? Wrow[ka + 1] : 0.0f;
    acc = wmma4(acc, a, bb);
  }
  const float bias = dtb[k * D_INNER + n];
  float* out = delta + ((size_t)(bk * D_INNER + n)) * LSEQ + mbase;
  #pragma unroll
  for (int i = 0; i < 8; ++i) {
    const float v = acc[i] + bias;
    out[i + 8 * half] = (v > 20.0f) ? v : __logf(1.0f + __expf(v));  // softplus
  }
}

// ---------------------------------------------------------------------------
// Chunked selective scan.  Sequence id seq = bk*D_INNER + d; 16 lanes = states.
// Pass A: per (seq, chunk) run recurrence from h0=0, produce chunk-local final
//         state h_part[n] and exact transition product T[n] = prod exp(dl*A_n).
// Pass B: per seq, combine 16 chunks serially: h0[c]=carry; carry=T*carry+h_part.
// Pass C: per (seq, chunk) re-run recurrence from exact h0, emit y.
// ---------------------------------------------------------------------------
__device__ __forceinline__ void scan_decode(int gid, int& chunk, int& d, int& bk) {
  chunk = gid % NCH;
  const int grp = gid / NCH;      // sequence
  d  = grp % D_INNER;
  bk = grp / D_INNER;
}

__global__ void k_scan_part(const float* __restrict__ xs2, const float* __restrict__ xdbl,
                            const float* __restrict__ delta, const float* __restrict__ A_logs,
                            float* __restrict__ hpart, float* __restrict__ tprod) {
  const int lane = threadIdx.x & 31;
  const int n    = lane & 15;
  const int gid  = (blockIdx.x * blockDim.x + threadIdx.x) >> 4;  // (seq,chunk)
  int chunk, d, bk;
  scan_decode(gid, chunk, d, bk);
  const int k = bk % 4, b = bk / 4;
  const int dir = k & 1, flip = k >> 1;

  const float Av = -__expf(A_logs[(k * D_INNER + d) * D_STATE + n]);
  const float* uB = xs2   + ((size_t)(b * 2 + dir) * D_INNER + d) * LSEQ;
  const float* dB = delta + ((size_t)bk * D_INNER + d) * LSEQ;
  const float* Bp = xdbl  + ((size_t)(bk * NDBL + DT_RANK) + n) * LSEQ;

  const int l0 = chunk * CHL;
  float h = 0.0f, T = 1.0f;
  for (int l = l0; l < l0 + CHL; l += 4) {
    const float4 bv = *(const float4*)(Bp + l);
    const float4 dv = *(const float4*)(dB + l);
    float4 uv;
    if (!flip) {
      uv = *(const float4*)(uB + l);
    } else {
      const float4 t = *(const float4*)(uB + (LSEQ - 4 - l));
      uv = make_float4(t.w, t.z, t.y, t.x);
    }
    const float dls[4] = {dv.x, dv.y, dv.z, dv.w};
    const float uls[4] = {uv.x, uv.y, uv.z, uv.w};
    const float bns[4] = {bv.x, bv.y, bv.z, bv.w};
    #pragma unroll
    for (int j = 0; j < 4; ++j) {
      const float e = __expf(dls[j] * Av);
      h = e * h + (dls[j] * uls[j]) * bns[j];
      T *= e;
    }
  }
  hpart[(size_t)gid * D_STATE + n] = h;
  tprod[(size_t)gid * D_STATE + n] = T;
}

__global__ void k_scan_comb(const float* __restrict__ hpart, const float* __restrict__ tprod,
                            float* __restrict__ h0arr) {
  const int lane = threadIdx.x & 31;
  const int n    = lane & 15;
  const int seq  = (blockIdx.x * blockDim.x + threadIdx.x) >> 4;  // 0..NSEQ-1
  float carry = 0.0f;
  #pragma unroll
  for (int c = 0; c < NCH; ++c) {
    const size_t idx = ((size_t)seq * NCH + c) * D_STATE + n;
    h0arr[idx] = carry;
    carry = tprod[idx] * carry + hpart[idx];
  }
}

__global__ void k_scan_final(const float* __restrict__ xs2, const float* __restrict__ xdbl,
                             const float* __restrict__ delta, const float* __restrict__ A_logs,
                             const float* __restrict__ Ds, const float* __restrict__ h0arr,
                             float* __restrict__ ys) {
  const int lane = threadIdx.x & 31;
  const int n    = lane & 15;
  const int gid  = (blockIdx.x * blockDim.x + threadIdx.x) >> 4;  // (seq,chunk)
  int chunk, d, bk;
  scan_decode(gid, chunk, d, bk);
  const int k = bk % 4, b = bk / 4;
  const int dir = k & 1, flip = k >> 1;

  const float Av = -__expf(A_logs[(k * D_INNER + d) * D_STATE + n]);
  const float Dp = Ds[k * D_INNER + d];
  const float* uB = xs2   + ((size_t)(b * 2 + dir) * D_INNER + d) * LSEQ;
  const float* dB = delta + ((size_t)bk * D_INNER + d) * LSEQ;
  const float* Bp = xdbl  + ((size_t)(bk * NDBL + DT_RANK) + n) * LSEQ;
  const float* Cp = xdbl  + ((size_t)(bk * NDBL + DT_RANK + D_STATE) + n) * LSEQ;
  float* yO = ys + ((size_t)bk * D_INNER + d) * LSEQ;

  const int l0 = chunk * CHL;
  float h = h0arr[(size_t)gid * D_STATE + n];
  for (int l = l0; l < l0 + CHL; l += 4) {
    const float4 bv = *(const float4*)(Bp + l);
    const float4 cv = *(const float4*)(Cp + l);
    const float4 dv = *(const float4*)(dB + l);
    float4 uv;
    if (!flip) {
      uv = *(const float4*)(uB + l);
    } else {
      const float4 t = *(const float4*)(uB + (LSEQ - 4 - l));
      uv = make_float4(t.w, t.z, t.y, t.x);
    }
    const float dls[4] = {dv.x, dv.y, dv.z, dv.w};
    const float uls[4] = {uv.x, uv.y, uv.z, uv.w};
    const float bns[4] = {bv.x, bv.y, bv.z, bv.w};
    const float cns[4] = {cv.x, cv.y, cv.z, cv.w};
    #pragma unroll
    for (int j = 0; j < 4; ++j) {
      const float dl = dls[j], ul = uls[j];
      h = __expf(dl * Av) * h + (dl * ul) * bns[j];
      float p = h * cns[j];
      p += __shfl_xor(p, 1, 32);
      p += __shfl_xor(p, 2, 32);
      p += __shfl_xor(p, 4, 32);
      p += __shfl_xor(p, 8, 32);
      if (n == 0) yO[l + j] = p + Dp * ul;
    }
  }
}

// ---------------------------------------------------------------------------
// Kernel 6: merge 4 directions + LayerNorm + SiLU gate.  One wave per pixel.
// ---------------------------------------------------------------------------
__global__ void k_merge(const float* __restrict__ ys, const float* __restrict__ z,
                        const float* __restrict__ nw, const float* __restrict__ nb,
                        float* __restrict__ ym) {
  const int lane = threadIdx.x & 31;
  const int pix  = (blockIdx.x * blockDim.x + threadIdx.x) >> 5;
  const int b  = pix / LSEQ;
  const int l0 = pix % LSEQ;
  const int h = l0 / WWID, w = l0 % WWID;
  const int l1 = w * HH + h;

  const size_t base = (size_t)b * 4 * D_INNER * LSEQ;
  float vals[6];
  float sum = 0.0f, sq = 0.0f;
  #pragma unroll
  for (int t = 0; t < 6; ++t) {
    const int d = lane + 32 * t;
    const float v = ys[base + ((size_t)(0 * D_INNER + d)) * LSEQ + l0]
                  + ys[base + ((size_t)(2 * D_INNER + d)) * LSEQ + (LSEQ - 1 - l0)]
                  + ys[base + ((size_t)(1 * D_INNER + d)) * LSEQ + l1]
                  + ys[base + ((size_t)(3 * D_INNER + d)) * LSEQ + (LSEQ - 1 - l1)];
    vals[t] = v;  sum += v;  sq += v * v;
  }
  #pragma unroll
  for (int m = 16; m >= 1; m >>= 1) {
    sum += __shfl_xor(sum, m, 32);
    sq  += __shfl_xor(sq,  m, 32);
  }
  const float mean = sum * (1.0f / D_INNER);
  const float var  = sq * (1.0f / D_INNER) - mean * mean;
  const float inv  = rsqrtf(var + 1e-5f);
  #pragma unroll
  for (int t = 0; t < 6; ++t) {
    const int d = lane + 32 * t;
    const float zy  = z[(size_t)pix * D_INNER + d];
    const float sil = zy * sigmoid_f(zy);
    ym[(size_t)pix * D_INNER + d] = ((vals[t] - mean) * inv * nw[d] + nb[d]) * sil;
  }
}

// ---------------------------------------------------------------------------
// Kernel 7: out_proj GEMM.  out[m][c] = sum_d ym[m][d] * W[c][d]
// M = 16384, N = 96 (6 tiles), K = 192.
// ---------------------------------------------------------------------------
__global__ void k_outproj(const float* __restrict__ ym, const float* __restrict__ opw,
                          float* __restrict__ out) {
  const int lane = threadIdx.x & 31;
  const int half = lane >> 4, id = lane & 15;
  int wave = (blockIdx.x * blockDim.x + threadIdx.x) >> 5;
  const int ntile = wave % 6;  wave /= 6;
  const int mtile = wave;                        // 0..1023
  const int mbase = mtile * 16, nbase = ntile * 16;

  const float* Arow = ym  + (size_t)(mbase + id) * D_INNER;
  const float* Brow = opw + (size_t)(nbase + id) * D_INNER;
  v8f acc = {};
  for (int kk = 0; kk < D_INNER; kk += 4) {
    const int ka = kk + 2 * half;
    v2f a, bb;
    a.x  = Arow[ka];  a.y  = Arow[ka + 1];
    bb.x = Brow[ka];  bb.y = Brow[ka + 1];
    acc = wmma4(acc, a, bb);
  }
  #pragma unroll
  for (int i = 0; i < 8; ++i) {
    const int m = mbase + i + 8 * half;
    out[(size_t)m * D_MODEL + nbase + id] = acc[i];
  }
}

// ---------------------------------------------------------------------------
extern "C" void kernel_launch(void* const* d_in, const int* in_sizes, int n_in,
                              void* d_out, int out_size, void* d_ws, size_t ws_size,
                              hipStream_t stream) {
  const float* x    = (const float*)d_in[0];
  const float* ipw  = (const float*)d_in[1];
  const float* cw   = (const float*)d_in[2];
  const float* cb   = (const float*)d_in[3];
  const float* xpw  = (const float*)d_in[4];
  const float* dtw  = (const float*)d_in[5];
  const float* dtb  = (const float*)d_in[6];
  const float* alog = (const float*)d_in[7];
  const float* dsv  = (const float*)d_in[8];
  const float* nw   = (const float*)d_in[9];
  const float* nb   = (const float*)d_in[10];
  const float* opw  = (const float*)d_in[11];
  float* out = (float*)d_out;

  // workspace partition (floats)
  float* ws = (float*)d_ws;
  const size_t SZ_XP   = (size_t)NPIX * D_INNER;                 //  3,145,728
  const size_t SZ_Z    = SZ_XP;
  const size_t SZ_XS2  = (size_t)BATCH * 2 * D_INNER * LSEQ;     //  6,291,456
  const size_t SZ_XDBL = (size_t)BATCH * KDIRS * NDBL * LSEQ;    //  2,490,368
  const size_t SZ_DLT  = (size_t)BATCH * KDIRS * D_INNER * LSEQ; // 12,582,912
  const size_t SZ_HP   = (size_t)NSEQ * NCH * D_STATE;           //    786,432
  float* xp    = ws;
  float* zbuf  = xp   + SZ_XP;
  float* xs2   = zbuf + SZ_Z;
  float* xdbl  = xs2  + SZ_XS2;
  float* delta = xdbl + SZ_XDBL;
  float* ysb   = delta + SZ_DLT;
  // xp region is dead between the conv kernel and the merge kernel; reuse it
  // for the chunked-scan scratch (3 * 786,432 floats = 9 MB <= 12 MB).
  float* hpart = xp;
  float* tprod = hpart + SZ_HP;
  float* h0arr = tprod + SZ_HP;
  float* ymrg  = xp;   // reused again after the scan, before kernel 6 writes it

  // 1) in_proj: 1024 Mtiles * 24 Ntiles = 24576 waves -> 3072 blocks of 8 waves
  k_in_proj<<<3072, 256, 0, stream>>>(x, ipw, xp, zbuf);
  // 2) depthwise conv + SiLU + directional copies: 16384*192 threads
  k_conv<<<(NPIX * D_INNER) / 256, 256, 0, stream>>>(xp, cw, cb, xs2);
  // 3) x_proj: 16 (b,k) * 256 Mtiles * 3 Ntiles = 12288 waves -> 1536 blocks
  k_xproj<<<1536, 256, 0, stream>>>(xs2, xpw, xdbl);
  // 4) dt_proj + softplus: 16 * 256 * 12 = 49152 waves -> 6144 blocks
  k_dtproj<<<6144, 256, 0, stream>>>(xdbl, dtw, dtb, delta);
  // 5) chunked scan:
  //    A) 3072 seq * 16 chunks * 16 lanes = 786432 threads -> 3072 blocks
  k_scan_part<<<3072, 256, 0, stream>>>(xs2, xdbl, delta, alog, hpart, tprod);
  //    B) 3072 seq * 16 lanes = 49152 threads -> 192 blocks (16 serial steps)
  k_scan_comb<<<192, 256, 0, stream>>>(hpart, tprod, h0arr);
  //    C) same shape as A, emits y
  k_scan_final<<<3072, 256, 0, stream>>>(xs2, xdbl, delta, alog, dsv, h0arr, ysb);
  // 6) merge + LayerNorm + gate: 16384 waves -> 2048 blocks
  k_merge<<<2048, 256, 0, stream>>>(ysb, zbuf, nw, nb, ymrg);
  // 7) out_proj: 1024 * 6 = 6144 waves -> 768 blocks
  k_outproj<<<768, 256, 0, stream>>>(ymrg, opw, out);
}